// GabrielGraphLoss_32220844654991
// MI455X (gfx1250) — compile-verified
//
#include <hip/hip_runtime.h>

typedef float v2f __attribute__((ext_vector_type(2)));
typedef float v8f __attribute__((ext_vector_type(8)));

#define N_NODES 768
#define MARGIN  0.01f
#define M_PAIRS (N_NODES * (N_NODES - 1) / 2)   /* 294528 */
#define TILES_M (M_PAIRS / 16)                  /* 18408  */
#define TILES_N (N_NODES / 16)                  /* 48     */
#define WAVES_PER_BLOCK 8
#define BLOCK_SIZE (WAVES_PER_BLOCK * 32)
#define GRID_BLOCKS (TILES_M / WAVES_PER_BLOCK) /* 2301   */
#define FLT_BIG 3.0e38f

__global__ void gabriel_zero_out(float* out) { out[0] = 0.0f; }

__device__ __forceinline__ int pair_row_offset(int i) {
    // number of pairs (a,b), a<b, with a < i  ==  i*(n-1) - i*(i-1)/2
    return i * (N_NODES - 1) - (i * (i - 1)) / 2;
}

// single-instruction max(x, 0) for finite x: v_med3_f32
__device__ __forceinline__ float relu1(float x) {
    return __builtin_amdgcn_fmed3f(x, 0.0f, FLT_BIG);
}

__global__ __launch_bounds__(BLOCK_SIZE)
void gabriel_loss_kernel(const float* __restrict__ pos, float* __restrict__ out) {
    const int lane = threadIdx.x & 31;
    const int wave = threadIdx.x >> 5;
    const int tile = blockIdx.x * WAVES_PER_BLOCK + wave;  // pair-tile [0, TILES_M)
    const int half = lane >> 4;    // 0 = lanes 0-15, 1 = lanes 16-31
    const int nloc = lane & 15;    // column within 16-wide tile

    // ---- decode pair p -> (i, j) with i < j (triu order; order irrelevant to mean)
    const int p = tile * 16 + nloc;
    const float twoN1 = 2.0f * (float)N_NODES - 1.0f;
    const float disc  = twoN1 * twoN1 - 8.0f * (float)p;   // >= 9, exact in fp32
    int i = (int)((twoN1 - __builtin_amdgcn_sqrtf(disc)) * 0.5f);
    if (i < 0) i = 0;
    if (i > N_NODES - 2) i = N_NODES - 2;
    while (pair_row_offset(i + 1) <= p) ++i;   // fixup (at most 1-2 steps)
    while (pair_row_offset(i) > p) --i;
    const int j = i + 1 + (p - pair_row_offset(i));

    // ---- pair data (all lanes compute their column's pair; halves duplicate work)
    const float pix = pos[2 * i + 0], piy = pos[2 * i + 1];
    const float pjx = pos[2 * j + 0], pjy = pos[2 * j + 1];
    const float mx = 0.5f * (pix + pjx), my = 0.5f * (piy + pjy);
    const float dx = pix - pjx, dy = piy - pjy;
    const float radius = 0.5f * __builtin_amdgcn_sqrtf(dx * dx + dy * dy) + MARGIN;
    const float msq = mx * mx + my * my;

    // B fragment (4x16 f32): lanes 0-15 hold rows K=0,1 ; lanes 16-31 rows K=2,3
    v2f bfrag;
    bfrag.x = half ? msq  : -2.0f * mx;   // K=0 : -2*mx   | K=2 : m^2
    bfrag.y = half ? 1.0f : -2.0f * my;   // K=1 : -2*my   | K=3 : 1

    float acc = 0.0f;

    for (int nt = 0; nt < TILES_N; ++nt) {
        const int node = nt * 16 + nloc;
        const float x = pos[2 * node + 0], y = pos[2 * node + 1];
        // A fragment (16x4 f32): lanes 0-15 hold K=0,1 = (x,y); lanes 16-31 K=2,3 = (1, x^2+y^2)
        v2f afrag;
        afrag.x = half ? 1.0f            : x;
        afrag.y = half ? (x * x + y * y) : y;

        v8f c = {};
        // D = A x B : D[k][p] = x^2 + m^2 - 2 x.m = dist^2
        c = __builtin_amdgcn_wmma_f32_16x16x4_f32(
                /*neg_a=*/false, afrag, /*neg_b=*/false, bfrag,
                /*c_mod=*/(short)0, c, /*reuse_a=*/false, /*reuse_b=*/false);

        // radius is lane-uniform across all 8 accumulator elements (N = lane&15)
#pragma unroll
        for (int v = 0; v < 8; ++v) {
            float dsq  = relu1(c[v]);                      // v_med3_f32
            float dist = __builtin_amdgcn_sqrtf(dsq);      // raw v_sqrt_f32
            float t    = relu1(radius - dist);             // v_sub + v_med3_f32
            acc = fmaf(t, t, acc);                         // v_fmac_f32
        }
    }

    // fold in the mean scale: 1 / (n * m)
    acc *= (float)(1.0 / ((double)N_NODES * (double)M_PAIRS));

    // ---- intra-wave reduction (wave32)
#pragma unroll
    for (int o = 16; o > 0; o >>= 1)
        acc += __shfl_xor(acc, o, 32);

    __shared__ float wsum[WAVES_PER_BLOCK];
    if (lane == 0) wsum[wave] = acc;
    __syncthreads();
    if (threadIdx.x == 0) {
        float s = 0.0f;
#pragma unroll
        for (int w = 0; w < WAVES_PER_BLOCK; ++w) s += wsum[w];
        atomicAdd(out, s);
    }
}

extern "C" void kernel_launch(void* const* d_in, const int* in_sizes, int n_in,
                              void* d_out, int out_size, void* d_ws, size_t ws_size,
                              hipStream_t stream) {
    (void)in_sizes; (void)n_in; (void)out_size; (void)d_ws; (void)ws_size;
    const float* pos = (const float*)d_in[0];   // node_pos [768,2] f32
    float* out = (float*)d_out;                 // scalar f32

    gabriel_zero_out<<<1, 1, 0, stream>>>(out);
    gabriel_loss_kernel<<<GRID_BLOCKS, BLOCK_SIZE, 0, stream>>>(pos, out);
}